// OscillatorBank_80255758893554
// MI455X (gfx1250) — compile-verified
//
#include <hip/hip_runtime.h>
#include <hip/hip_bf16.h>
#include <math.h>
#include <stdint.h>

#define N_B   16
#define N_T   500
#define N_H   100
#define HOP   64
#define SR_F  44100.0f
#define NYQ_F 22050.0f
#define TWO_PI_F 6.28318530717958647692f

// ---------------------------------------------------------------------------
// Kernel A: per-(b,t,h) harmonic frequency (rad/sample) and masked, pre-scaled
// amplitude (folds loudness and the 1/N_HARMONICS normalization).
// pow(h+1, 1+stretch) with h+1 in [1,100] (always positive, finite) is safely
// computed on the native TRANS path via the AMDGPU intrinsics:
//   v_log_f32 (log2) + v_exp_f32 (2^x) — no special-case ladder.
// ---------------------------------------------------------------------------
__global__ void osc_prep(const float* __restrict__ f0,
                         const float* __restrict__ loud,
                         const float* __restrict__ amps,
                         const float* __restrict__ stretch,
                         float* __restrict__ w_out,
                         float* __restrict__ a_out) {
  int idx = blockIdx.x * blockDim.x + threadIdx.x;
  if (idx >= N_B * N_T * N_H) return;
  int h  = idx % N_H;          // harmonic index 0..99  (harmonic number h+1)
  int bt = idx / N_H;          // flattened (b,t)
  float st = stretch[bt];
  float fv = f0[bt];
  float lv = loud[bt];
  float l2h  = __builtin_amdgcn_logf((float)(h + 1));        // v_log_f32
  float harm = __builtin_amdgcn_exp2f((1.0f + st) * l2h) * fv; // v_exp_f32; Hz
  float w = harm * (TWO_PI_F / SR_F);                  // radians per sample
  float a = (harm > NYQ_F) ? 0.0f : amps[idx];         // Nyquist mask
  w_out[idx] = w;
  a_out[idx] = a * lv * (1.0f / (float)N_H);           // fold loudness & 1/H
}

// ---------------------------------------------------------------------------
// Kernel B: serial scan over frames. base[b,t,h] = frame-start phase in
// [0, 2pi). f64 accumulator with branch-free range reduction (mul/floor/fma)
// instead of the fmod libcall; per-frame increment is <= ~91 rad so the
// reduction is exact to f64 precision.
// ---------------------------------------------------------------------------
__global__ void osc_scan(const float* __restrict__ w_in,
                         float* __restrict__ base_out) {
  int i = blockIdx.x * blockDim.x + threadIdx.x;
  if (i >= N_B * N_H) return;
  int b = i / N_H;
  int h = i % N_H;
  const double two_pi  = 6.283185307179586476925286766559;
  const double inv2pi  = 0.159154943091895335768883763373;
  double phase = 0.0;                               // always in [0, 2pi)
  for (int t = 0; t < N_T; ++t) {
    int idx = (b * N_T + t) * N_H + h;
    base_out[idx] = (float)phase;                   // phase at frame start
    phase += 64.0 * (double)w_in[idx];              // advance one hop
    phase -= floor(phase * inv2pi) * two_pi;        // reduce to [0, 2pi)
  }
}

// ---------------------------------------------------------------------------
// gfx1250 async copy: per-lane 8B global -> LDS, tracked by ASYNCcnt.
// lds_off is the wave-relative LDS byte offset (generic-pointer low 32 bits).
// ---------------------------------------------------------------------------
__device__ __forceinline__ void async_b64_to_lds(unsigned lds_off, const void* g) {
  asm volatile("global_load_async_to_lds_b64 %0, %1, off"
               :: "v"(lds_off), "v"((unsigned long long)(uintptr_t)g)
               : "memory");
}

// ---------------------------------------------------------------------------
// Kernel C: one 64-thread block (2 wave32s) per frame; 8000 blocks total.
// Stage w/base/amp (3 x 100 f32) into LDS via async-to-LDS, then lane j
// evaluates sample j of the frame: acc = sum_h a[h]*sin(base[h]+(j+1)w[h]).
// All lanes read the same h each iteration -> conflict-free LDS broadcast
// (unroll-4 body compiles to three ds_load_b128 per 4 harmonics).
// __sinf lowers to the native v_sin_f32 TRANS path (co-executes with VALU);
// |phase| <= ~97 rad stays inside the fast-path domain.
// ---------------------------------------------------------------------------
__global__ void __launch_bounds__(HOP)
osc_synth(const float* __restrict__ w_in,
          const float* __restrict__ base_in,
          const float* __restrict__ a_in,
          float* __restrict__ out) {
  __shared__ float smem[3 * N_H + 4];
  const int frame = blockIdx.x;             // b*N_T + t
  const int j     = threadIdx.x;            // 0..63 sample within frame
  const int fb    = frame * N_H;

  const unsigned lds0 = (unsigned)(uintptr_t)smem;  // LDS byte offset
  if (j < N_H / 2) {                         // 50 lanes x 8B = 400B per array
    async_b64_to_lds(lds0 +        j * 8u, w_in    + fb + j * 2);
    async_b64_to_lds(lds0 + 400u + j * 8u, base_in + fb + j * 2);
    async_b64_to_lds(lds0 + 800u + j * 8u, a_in    + fb + j * 2);
  }
  asm volatile("s_wait_asynccnt 0" ::: "memory");   // my wave's copies landed
  __syncthreads();                                  // publish LDS to both waves

  const float* sw = smem;            // byte 0
  const float* sb = smem + N_H;      // byte 400
  const float* sa = smem + 2 * N_H;  // byte 800

  const float jp1 = (float)(j + 1);
  float acc = 0.0f;
#pragma unroll 4
  for (int h = 0; h < N_H; ++h) {
    float ph = fmaf(jp1, sw[h], sb[h]);   // base + (j+1)*w
    acc = fmaf(sa[h], __sinf(ph), acc);
  }
  out[frame * HOP + j] = acc;              // coalesced: b*32000 + t*64 + j
}

// ---------------------------------------------------------------------------
extern "C" void kernel_launch(void* const* d_in, const int* in_sizes, int n_in,
                              void* d_out, int out_size, void* d_ws, size_t ws_size,
                              hipStream_t stream) {
  (void)in_sizes; (void)n_in; (void)out_size; (void)ws_size;
  const float* f0      = (const float*)d_in[0];
  const float* loud    = (const float*)d_in[1];
  const float* amps    = (const float*)d_in[2];
  const float* stretch = (const float*)d_in[3];
  float* out = (float*)d_out;

  const int NE = N_B * N_T * N_H;          // 800000
  float* w    = (float*)d_ws;              // [B,T,H] rad/sample
  float* base = w    + NE;                 // [B,T,H] frame-start phase [0,2pi)
  float* aa   = base + NE;                 // [B,T,H] masked amp * loud / H
                                           // total 9.6 MB of workspace

  osc_prep<<<(NE + 255) / 256, 256, 0, stream>>>(f0, loud, amps, stretch, w, aa);

  const int NS = N_B * N_H;                // 1600 scan threads
  osc_scan<<<(NS + 255) / 256, 256, 0, stream>>>(w, base);

  osc_synth<<<N_B * N_T, HOP, 0, stream>>>(w, base, aa, out);
}